// DTW_82085414961776
// MI455X (gfx1250) — compile-verified
//
#include <hip/hip_runtime.h>
#include <hip/hip_bf16.h>

typedef __attribute__((ext_vector_type(2))) float v2f;
typedef __attribute__((ext_vector_type(8))) float v8f;

#define BATCH 64
#define L     1024   // L1 == L2 == 1024

// ---------------------------------------------------------------------------
// Linear layer via V_WMMA_F32_16X16X4_F32:  P[M,N] = X[M,K] @ W[N,K]^T + bias
// One wave per 16x16 output tile; K-loop in steps of 4.
//
// A frag (16x4 f32, 2 VGPRs/lane):  lanes 0-15 -> M=lane, VGPR{0,1}=K{0,1};
//                                   lanes 16-31 -> M=lane-16, VGPR{0,1}=K{2,3}
// B frag (4x16 f32, 2 VGPRs/lane):  mirrored: n = lane&15, k = (lane>>4)*2 + v
//   B[k][n] = W[n][k], so both frags are contiguous float2 loads along K.
// C/D (16x16 f32, 8 VGPRs): n = lane&15, m = v + (lane>>4)*8.
// ---------------------------------------------------------------------------
__global__ __launch_bounds__(256)
void linear_wmma_kernel(const float* __restrict__ X, const float* __restrict__ W,
                        const float* __restrict__ bias, float* __restrict__ P,
                        int M, int N, int K) {
  const int gtid  = blockIdx.x * blockDim.x + threadIdx.x;
  const int wave  = gtid >> 5;
  const int lane  = threadIdx.x & 31;
  const int tilesN = N >> 4;
  const int tm = wave / tilesN;          // output tile row (M/16)
  const int tn = wave % tilesN;          // output tile col (N/16)
  if (tm * 16 >= M) return;

  const int m0   = tm << 4;
  const int n0   = tn << 4;
  const int half = lane >> 4;            // 0: lanes 0-15, 1: lanes 16-31
  const int l15  = lane & 15;

  const float* arow = X + (size_t)(m0 + l15) * K + half * 2;  // A[m, k..k+1]
  const float* brow = W + (size_t)(n0 + l15) * K + half * 2;  // B[k][n] = W[n][k]

  v8f acc = {};
  for (int k = 0; k < K; k += 4) {
    v2f afrag = *(const v2f*)(arow + k);
    v2f bfrag = *(const v2f*)(brow + k);
    // 8 args: (neg_a, A, neg_b, B, c_mod, C, reuse_a, reuse_b)
    acc = __builtin_amdgcn_wmma_f32_16x16x4_f32(
        false, afrag, false, bfrag, (short)0, acc, false, false);
  }

  const float bn = bias[n0 + l15];
#pragma unroll
  for (int v = 0; v < 8; ++v) {
    const int m = v + half * 8;
    P[(size_t)(m0 + m) * N + n0 + l15] = acc[v] + bn;
  }
}

// ---------------------------------------------------------------------------
// DTW accumulated-cost DP, one workgroup per batch element.
//   R[i,j] = (ap[i]-bp[j])^2 + min(R[i-1,j-1], R[i-1,j], R[i,j-1])
// Anti-diagonal wavefront: thread i owns row i; diagonal d holds cells (i, d-i).
// 3-deep rolling diagonal buffer in LDS; invalid cells carry +INF so the
// borders fall out automatically. One barrier per diagonal (2L-1 = 2047 steps).
// Cost matrix (256 MB) is never materialized: ap/bp rows live in LDS (8 KB).
// ---------------------------------------------------------------------------
__global__ __launch_bounds__(1024)
void dtw_kernel(const float* __restrict__ ap, const float* __restrict__ bp,
                float* __restrict__ out) {
  const int b = blockIdx.x;
  const int i = threadIdx.x;            // row index, 0..L-1

  __shared__ float sa[L];
  __shared__ float sb[L];
  __shared__ float diag[3][L];          // rolling anti-diagonals d, d-1, d-2

  sa[i] = ap[(size_t)b * L + i];
  sb[i] = bp[(size_t)b * L + i];
  __syncthreads();

  const float INF = __builtin_inff();
  float r_final = 0.0f;

  for (int d = 0; d < 2 * L - 1; ++d) {
    const int j = d - i;
    float r = INF;
    if (j >= 0 && j < L) {
      float c = sa[i] - sb[j];
      c *= c;
      float best;
      if (d == 0) {
        best = 0.0f;                                   // R[0,0] = cost[0,0]
      } else {
        const float* Dp  = diag[(d + 2) % 3];          // diagonal d-1
        const float* Dp2 = diag[(d + 1) % 3];          // diagonal d-2
        const float up   = (i > 0)          ? Dp[i - 1]  : INF;  // R[i-1,j]
        const float left =                    Dp[i];             // R[i,j-1] (INF if j-1<0)
        const float dg   = (i > 0 && j > 0) ? Dp2[i - 1] : INF;  // R[i-1,j-1]
        best = fminf(dg, fminf(up, left));
      }
      r = c + best;
    }
    diag[d % 3][i] = r;                 // writes buffer d-3: disjoint from reads
    if (d == 2 * L - 2) r_final = r;    // only thread i==L-1 is valid here
    __syncthreads();
  }

  if (i == L - 1) out[b] = r_final;     // R[L-1, L-1]
}

// ---------------------------------------------------------------------------
extern "C" void kernel_launch(void* const* d_in, const int* in_sizes, int n_in,
                              void* d_out, int out_size, void* d_ws, size_t ws_size,
                              hipStream_t stream) {
  const float* a  = (const float*)d_in[0];   // [B, L]
  const float* b  = (const float*)d_in[1];   // [B, L]
  const float* W1 = (const float*)d_in[2];   // [L, L]
  const float* b1 = (const float*)d_in[3];   // [L]
  const float* W2 = (const float*)d_in[4];   // [L, L]
  const float* b2 = (const float*)d_in[5];   // [L]
  float* out = (float*)d_out;                // [B]

  float* ap = (float*)d_ws;                  // [B, L] projected seq1
  float* bp = ap + (size_t)BATCH * L;        // [B, L] projected seq2

  // (B/16)*(L/16) = 4*64 = 256 waves = 8192 threads per GEMM
  const int gemm_threads = 256;
  const int gemm_blocks  = (BATCH / 16) * (L / 16) * 32 / gemm_threads;  // 32

  linear_wmma_kernel<<<gemm_blocks, gemm_threads, 0, stream>>>(
      a, W1, b1, ap, BATCH, L, L);
  linear_wmma_kernel<<<gemm_blocks, gemm_threads, 0, stream>>>(
      b, W2, b2, bp, BATCH, L, L);

  dtw_kernel<<<BATCH, L, 0, stream>>>(ap, bp, out);
}